// GAT_LSTM_24489903521814
// MI455X (gfx1250) — compile-verified
//
#include <hip/hip_runtime.h>

// ---------------------------------------------------------------------------
// GAT(2 layers, fixed 17-node skeleton graph) -> BiLSTM x2 -> MLP head
// All GEMMs via V_WMMA_F32_16X16X4_F32 (full fp32 matrix pipe on gfx1250).
// 32x32 macro-tile per wave (2x2 WMMA tiles), register double-buffered loads,
// last main-loop iteration peeled so prefetch addressing strength-reduces.
// ---------------------------------------------------------------------------

#define SEQ   90
#define KP    17
#define FD    2
#define BATCH 64
#define NGR   (BATCH * SEQ)        // 5760 graphs
#define NN    (NGR * KP)           // 97920 nodes
#define HID   512
#define GATE  2048
#define D_IN1 1122                 // KP*(64+FD)
#define D_IN2 1024

typedef float v2f __attribute__((ext_vector_type(2)));
typedef float v8f __attribute__((ext_vector_type(8)));

__device__ __forceinline__ v8f wmma_f32(v2f a, v2f b, v8f c) {
    // (neg_a, A, neg_b, B, c_mod, C, reuse_a, reuse_b)
    return __builtin_amdgcn_wmma_f32_16x16x4_f32(false, a, false, b, (short)0, c,
                                                 false, false);
}

__device__ __forceinline__ float sigmoidf_(float x) { return 1.0f / (1.0f + __expf(-x)); }

// Fixed skeleton adjacency (undirected neighbors per node; self-loop added in kernel)
__constant__ int NBR_OFF[18] = {0,2,5,8,10,12,16,20,22,24,25,26,29,32,34,36,37,38};
__constant__ int NBR[38] = {
    1,2, 2,0,3, 1,0,4, 1,5, 2,6, 11,6,7,3, 12,5,8,4, 5,9, 6,10, 7, 8,
    13,12,5, 14,11,6, 15,11, 16,12, 13, 14
};

// ---------------------------------------------------------------------------
// fp32 WMMA GEMM:  C[M,N] = A[M,K] * B[N,K]^T (+bias, +relu)
// Requirements (all call sites satisfy): M % 32 == 0, K >= 8, lda/ldb even.
// One wave owns a 32x32 output tile = 2x2 WMMA tiles. Fragments are contiguous
// f32 pairs -> one b64 load each; next-k fragments prefetched while current
// WMMAs execute; final main iteration peeled (no clamped addressing in loop).
// ---------------------------------------------------------------------------
__global__ void gemm_wmma_f32(const float* __restrict__ A, int lda,
                              const float* __restrict__ Bm, int ldb,
                              const float* __restrict__ bias,
                              float* __restrict__ C, int ldc,
                              int M, int N, int K, int act) {
    int wave = (blockIdx.x * blockDim.x + threadIdx.x) >> 5;
    int lane = threadIdx.x & 31;
    int tilesN = (N + 31) >> 5;
    int tilesM = M >> 5;
    if (wave >= tilesM * tilesN) return;           // wave-uniform
    int tm = wave / tilesN, tn = wave % tilesN;
    int m0 = tm << 5, n0 = tn << 5;
    int half = lane >> 4, l = lane & 15;
    int kofs = half << 1;                          // lane's K sub-offset (0 or 2)

    const float* A0 = A + (size_t)(m0 + l) * lda + kofs;
    const float* A1 = A0 + (size_t)16 * lda;
    int bn0 = n0 + l;      if (bn0 > N - 1) bn0 = N - 1;   // clamp edge-tile loads
    int bn1 = n0 + 16 + l; if (bn1 > N - 1) bn1 = N - 1;
    const float* B0 = Bm + (size_t)bn0 * ldb + kofs;
    const float* B1 = Bm + (size_t)bn1 * ldb + kofs;

    v8f acc00 = {0.f,0.f,0.f,0.f,0.f,0.f,0.f,0.f};
    v8f acc01 = acc00, acc10 = acc00, acc11 = acc00;

    int Kmain = K & ~3;                            // >= 8 at all call sites
    v2f a0 = *(const v2f*)A0;
    v2f a1 = *(const v2f*)A1;
    v2f b0 = *(const v2f*)B0;
    v2f b1 = *(const v2f*)B1;
    for (int k0 = 0; k0 < Kmain - 4; k0 += 4) {    // plain induction -> ptr increments
        v2f a0n = *(const v2f*)(A0 + k0 + 4);
        v2f a1n = *(const v2f*)(A1 + k0 + 4);
        v2f b0n = *(const v2f*)(B0 + k0 + 4);
        v2f b1n = *(const v2f*)(B1 + k0 + 4);
        acc00 = wmma_f32(a0, b0, acc00);
        acc01 = wmma_f32(a0, b1, acc01);
        acc10 = wmma_f32(a1, b0, acc10);
        acc11 = wmma_f32(a1, b1, acc11);
        a0 = a0n; a1 = a1n; b0 = b0n; b1 = b1n;
    }
    acc00 = wmma_f32(a0, b0, acc00);               // peeled last main iteration
    acc01 = wmma_f32(a0, b1, acc01);
    acc10 = wmma_f32(a1, b0, acc10);
    acc11 = wmma_f32(a1, b1, acc11);

    if (K & 3) {                                   // K tail (e.g. K=1122 -> 2)
        int ka = Kmain + kofs, kb = ka + 1;
        int kaC = ka < K ? ka : K - 1;             // clamped, unconditional loads
        int kbC = kb < K ? kb : K - 1;
        const float* A0b = A + (size_t)(m0 + l) * lda;
        const float* A1b = A0b + (size_t)16 * lda;
        const float* B0b = Bm + (size_t)bn0 * ldb;
        const float* B1b = Bm + (size_t)bn1 * ldb;
        float va0 = A0b[kaC], wa0 = A0b[kbC];
        float va1 = A1b[kaC], wa1 = A1b[kbC];
        float vb0 = B0b[kaC], wb0 = B0b[kbC];
        float vb1 = B1b[kaC], wb1 = B1b[kbC];
        v2f at0, at1, bt0, bt1;
        at0.x = ka < K ? va0 : 0.f; at0.y = kb < K ? wa0 : 0.f;
        at1.x = ka < K ? va1 : 0.f; at1.y = kb < K ? wa1 : 0.f;
        bt0.x = ka < K ? vb0 : 0.f; bt0.y = kb < K ? wb0 : 0.f;
        bt1.x = ka < K ? vb1 : 0.f; bt1.y = kb < K ? wb1 : 0.f;
        acc00 = wmma_f32(at0, bt0, acc00);
        acc01 = wmma_f32(at0, bt1, acc01);
        acc10 = wmma_f32(at1, bt0, acc10);
        acc11 = wmma_f32(at1, bt1, acc11);
    }
    // D layout: VGPR j -> row j + 8*half ; col = lane&15
    int cn0 = n0 + l, cn1 = n0 + 16 + l;
    if (cn0 < N) {
        float bv = bias ? bias[cn0] : 0.f;
        #pragma unroll
        for (int j = 0; j < 8; ++j) {
            int r0 = m0 + j + half * 8, r1 = r0 + 16;
            float v0 = acc00[j] + bv, v1 = acc10[j] + bv;
            if (act) { v0 = fmaxf(v0, 0.f); v1 = fmaxf(v1, 0.f); }
            C[(size_t)r0 * ldc + cn0] = v0;
            C[(size_t)r1 * ldc + cn0] = v1;
        }
    }
    if (cn1 < N) {
        float bv = bias ? bias[cn1] : 0.f;
        #pragma unroll
        for (int j = 0; j < 8; ++j) {
            int r0 = m0 + j + half * 8, r1 = r0 + 16;
            float v0 = acc01[j] + bv, v1 = acc11[j] + bv;
            if (act) { v0 = fmaxf(v0, 0.f); v1 = fmaxf(v1, 0.f); }
            C[(size_t)r0 * ldc + cn1] = v0;
            C[(size_t)r1 * ldc + cn1] = v1;
        }
    }
}

// ---------------------------------------------------------------------------
// LSTM recurrent step: Zfull[d][b][n] = Zx[d][(b*90 + t_d)][n] + h[d][b]·Whh[d][n]
// M=64, N=2048, K=512; 32x32 tile per wave; both directions (blockIdx.y).
// Launch: dim3(16, 2) x 256 threads -> 128 waves per direction.
// ---------------------------------------------------------------------------
__global__ void lstm_step_gemm(const float* __restrict__ Zx,
                               const float* __restrict__ Whh,
                               const float* __restrict__ hprev,
                               float* __restrict__ Zfull, int t) {
    int dir  = blockIdx.y;
    int wave = blockIdx.x * 8 + (threadIdx.x >> 5);   // 0..127
    int lane = threadIdx.x & 31;
    int tm = wave >> 6, tn = wave & 63;               // 2 x 64 tiles
    int m0 = tm << 5, n0 = tn << 5;
    int half = lane >> 4, l = lane & 15;
    int kofs = half << 1;

    const float* A0 = hprev + (size_t)dir * BATCH * HID + (size_t)(m0 + l) * HID + kofs;
    const float* A1 = A0 + (size_t)16 * HID;
    const float* B0 = Whh + (size_t)dir * GATE * HID + (size_t)(n0 + l) * HID + kofs;
    const float* B1 = B0 + (size_t)16 * HID;

    v8f acc00 = {0.f,0.f,0.f,0.f,0.f,0.f,0.f,0.f};
    v8f acc01 = acc00, acc10 = acc00, acc11 = acc00;

    v2f a0 = *(const v2f*)A0;
    v2f a1 = *(const v2f*)A1;
    v2f b0 = *(const v2f*)B0;
    v2f b1 = *(const v2f*)B1;
    for (int k0 = 0; k0 < HID - 4; k0 += 4) {
        v2f a0n = *(const v2f*)(A0 + k0 + 4);
        v2f a1n = *(const v2f*)(A1 + k0 + 4);
        v2f b0n = *(const v2f*)(B0 + k0 + 4);
        v2f b1n = *(const v2f*)(B1 + k0 + 4);
        acc00 = wmma_f32(a0, b0, acc00);
        acc01 = wmma_f32(a0, b1, acc01);
        acc10 = wmma_f32(a1, b0, acc10);
        acc11 = wmma_f32(a1, b1, acc11);
        a0 = a0n; a1 = a1n; b0 = b0n; b1 = b1n;
    }
    acc00 = wmma_f32(a0, b0, acc00);
    acc01 = wmma_f32(a0, b1, acc01);
    acc10 = wmma_f32(a1, b0, acc10);
    acc11 = wmma_f32(a1, b1, acc11);

    int t_d = dir ? (SEQ - 1 - t) : t;
    const float* Zxd = Zx + (size_t)dir * NGR * GATE;
    float* Zfd = Zfull + (size_t)dir * BATCH * GATE;
    int cn0 = n0 + l, cn1 = n0 + 16 + l;
    #pragma unroll
    for (int j = 0; j < 8; ++j) {
        int r0 = m0 + j + half * 8, r1 = r0 + 16;
        Zfd[(size_t)r0 * GATE + cn0] = acc00[j] + Zxd[(size_t)(r0 * SEQ + t_d) * GATE + cn0];
        Zfd[(size_t)r0 * GATE + cn1] = acc01[j] + Zxd[(size_t)(r0 * SEQ + t_d) * GATE + cn1];
        Zfd[(size_t)r1 * GATE + cn0] = acc10[j] + Zxd[(size_t)(r1 * SEQ + t_d) * GATE + cn0];
        Zfd[(size_t)r1 * GATE + cn1] = acc11[j] + Zxd[(size_t)(r1 * SEQ + t_d) * GATE + cn1];
    }
}

// gates i,f,g,o -> c,h update; h -> ys[(b*90 + t_d)][dir*512 + j]
__global__ void lstm_cell(const float* __restrict__ Zfull,
                          float* __restrict__ hbuf, float* __restrict__ cbuf,
                          float* __restrict__ ys, int t) {
    int idx = blockIdx.x * blockDim.x + threadIdx.x;            // 2*64*512
    if (idx >= 2 * BATCH * HID) return;
    int dir = idx >> 15, r = idx & 32767;
    int b = r >> 9, j = r & 511;
    const float* z = Zfull + (size_t)dir * BATCH * GATE + (size_t)b * GATE;
    float zi = z[j], zf = z[HID + j], zg = z[2 * HID + j], zo = z[3 * HID + j];
    float c = cbuf[idx];
    c = sigmoidf_(zf) * c + sigmoidf_(zi) * tanhf(zg);
    float h = sigmoidf_(zo) * tanhf(c);
    cbuf[idx] = c; hbuf[idx] = h;
    int t_d = dir ? (SEQ - 1 - t) : t;
    ys[(size_t)(b * SEQ + t_d) * 1024 + dir * HID + j] = h;
}

// ---------------------------------------------------------------------------
// GAT kernels
// ---------------------------------------------------------------------------
__global__ void gat1_lin(const float* __restrict__ x, const float* __restrict__ W,
                         float* __restrict__ h) {
    int idx = blockIdx.x * blockDim.x + threadIdx.x;
    if (idx >= NN * 64) return;
    int i = idx >> 6, f = idx & 63;
    h[idx] = x[i * 2 + 0] * W[f] + x[i * 2 + 1] * W[64 + f];
}

__global__ void gat_scores(const float* __restrict__ h,
                           const float* __restrict__ asrc, const float* __restrict__ adst,
                           float* __restrict__ sas, float* __restrict__ sad) {
    int i = blockIdx.x * blockDim.x + threadIdx.x;
    if (i >= NN) return;
    const float* hp = h + (size_t)i * 64;
    float a = 0.f, d = 0.f;
    #pragma unroll
    for (int f = 0; f < 64; ++f) { float v = hp[f]; a += v * asrc[f]; d += v * adst[f]; }
    sas[i] = a; sad[i] = d;
}

// attention softmax (self-loop + <=4 neighbors) + bias + BN + ReLU, fused
__global__ void gat_aggr(const float* __restrict__ h,
                         const float* __restrict__ sas, const float* __restrict__ sad,
                         const float* __restrict__ bvec,
                         const float* __restrict__ g, const float* __restrict__ bb,
                         const float* __restrict__ bm, const float* __restrict__ bvv,
                         float* __restrict__ out) {
    int idx = blockIdx.x * blockDim.x + threadIdx.x;
    if (idx >= NN * 64) return;
    int i = idx >> 6, ch = idx & 63;
    int dl = i % KP, base = i - dl;
    float ad = sad[i];
    int o0 = NBR_OFF[dl], deg = NBR_OFF[dl + 1] - o0;

    float e[5], hv[5];
    float t0 = sas[i] + ad;                            // self loop
    e[0]  = t0 < 0.f ? 0.2f * t0 : t0;
    hv[0] = h[(size_t)i * 64 + ch];
    for (int n = 0; n < deg; ++n) {
        int s = base + NBR[o0 + n];
        float tt = sas[s] + ad;
        e[n + 1]  = tt < 0.f ? 0.2f * tt : tt;
        hv[n + 1] = h[(size_t)s * 64 + ch];
    }
    int cnt = deg + 1;
    float m = e[0];
    for (int n = 1; n < cnt; ++n) m = fmaxf(m, e[n]);
    float den = 0.f, num = 0.f;
    for (int n = 0; n < cnt; ++n) { float w = __expf(e[n] - m); den += w; num += w * hv[n]; }
    float val = num / den + bvec[ch];
    val = (val - bm[ch]) * (g[ch] * rsqrtf(bvv[ch] + 1e-5f)) + bb[ch];
    out[idx] = fmaxf(val, 0.f);
}

// [h2 | x] concat reshaped to [B*T, 1122]
__global__ void pack_emb(const float* __restrict__ h2, const float* __restrict__ x,
                         float* __restrict__ emb) {
    int idx = blockIdx.x * blockDim.x + threadIdx.x;
    if (idx >= NN * 66) return;
    int i = idx / 66, c = idx % 66;
    int bt = i / KP, kp = i % KP;
    float v = (c < 64) ? h2[(size_t)i * 64 + c] : x[i * 2 + (c - 64)];
    emb[(size_t)bt * D_IN1 + kp * 66 + c] = v;
}

__global__ void transpose64(const float* __restrict__ W, float* __restrict__ WT) {
    int idx = blockIdx.x * blockDim.x + threadIdx.x;   // 4096
    if (idx >= 64 * 64) return;
    int n = idx >> 6, k = idx & 63;
    WT[n * 64 + k] = W[k * 64 + n];
}

__global__ void zero_f32(float* __restrict__ p, int n) {
    int i = blockIdx.x * blockDim.x + threadIdx.x;
    if (i < n) p[i] = 0.f;
}

// ---------------------------------------------------------------------------
// Host side
// ---------------------------------------------------------------------------
static inline void launch_gemm(hipStream_t s, const float* A, int lda,
                               const float* B, int ldb, const float* bias,
                               float* C, int ldc, int M, int N, int K, int act) {
    int tiles = (M / 32) * ((N + 31) / 32);
    int blocks = (tiles + 7) / 8;                       // 8 waves / 256-thread block
    gemm_wmma_f32<<<blocks, 256, 0, s>>>(A, lda, B, ldb, bias, C, ldc, M, N, K, act);
}

extern "C" void kernel_launch(void* const* d_in, const int* in_sizes, int n_in,
                              void* d_out, int out_size, void* d_ws, size_t ws_size,
                              hipStream_t stream) {
    const float* x        = (const float*)d_in[0];
    /* d_in[1] edge_index unused: skeleton adjacency is constant */
    const float* gat1_W   = (const float*)d_in[2];
    const float* gat1_as  = (const float*)d_in[3];
    const float* gat1_ad  = (const float*)d_in[4];
    const float* gat1_b   = (const float*)d_in[5];
    const float* gat2_W   = (const float*)d_in[6];
    const float* gat2_as  = (const float*)d_in[7];
    const float* gat2_ad  = (const float*)d_in[8];
    const float* gat2_b   = (const float*)d_in[9];
    const float* bn1_g = (const float*)d_in[10], *bn1_b = (const float*)d_in[11];
    const float* bn1_m = (const float*)d_in[12], *bn1_v = (const float*)d_in[13];
    const float* bn2_g = (const float*)d_in[14], *bn2_b = (const float*)d_in[15];
    const float* bn2_m = (const float*)d_in[16], *bn2_v = (const float*)d_in[17];
    const float* l1_Wih = (const float*)d_in[18];
    const float* l1_Whh = (const float*)d_in[19];
    const float* l1_b   = (const float*)d_in[20];
    const float* l2_Wih = (const float*)d_in[21];
    const float* l2_Whh = (const float*)d_in[22];
    const float* l2_b   = (const float*)d_in[23];
    const float* hW1 = (const float*)d_in[24], *hb1 = (const float*)d_in[25];
    const float* hW2 = (const float*)d_in[26], *hb2 = (const float*)d_in[27];
    float* out = (float*)d_out;
    float* ws  = (float*)d_ws;

    // workspace layout (floats), aggressively aliased (~173 MB total)
    const size_t OFF_HA  = 0;                              // NN*64 ; later head1
    const size_t OFF_HB  = OFF_HA  + (size_t)NN * 64;      // NN*64 ; later ys2
    const size_t OFF_SAS = OFF_HB  + (size_t)NN * 64;
    const size_t OFF_SAD = OFF_SAS + NN;
    const size_t OFF_EMB = OFF_SAD + NN;                   // 5760*1122 ; later ys1
    const size_t OFF_ZX  = OFF_EMB + (size_t)NGR * D_IN1;  // 2*5760*2048
    const size_t OFF_ZF  = OFF_ZX  + (size_t)2 * NGR * GATE;
    const size_t OFF_HST = OFF_ZF  + (size_t)2 * BATCH * GATE;
    const size_t OFF_CST = OFF_HST + (size_t)2 * BATCH * HID;
    const size_t OFF_WT  = OFF_CST + (size_t)2 * BATCH * HID;   // 64*64 transposed gat2_W

    float* hA   = ws + OFF_HA;
    float* hB   = ws + OFF_HB;
    float* sas  = ws + OFF_SAS;
    float* sad  = ws + OFF_SAD;
    float* emb  = ws + OFF_EMB;
    float* Zx   = ws + OFF_ZX;
    float* Zf   = ws + OFF_ZF;
    float* hst  = ws + OFF_HST;
    float* cst  = ws + OFF_CST;
    float* wt   = ws + OFF_WT;
    float* ys1  = emb;    // emb dead after Zx-layer1 GEMM
    float* ys2  = hB;     // hB dead after pack_emb
    float* hd1  = hA;     // hA dead after pack_emb

    const int TPB = 256;

    // ---- GAT layer 1 ----
    gat1_lin<<<(NN * 64 + TPB - 1) / TPB, TPB, 0, stream>>>(x, gat1_W, hA);
    gat_scores<<<(NN + TPB - 1) / TPB, TPB, 0, stream>>>(hA, gat1_as, gat1_ad, sas, sad);
    gat_aggr<<<(NN * 64 + TPB - 1) / TPB, TPB, 0, stream>>>(hA, sas, sad, gat1_b,
                                                            bn1_g, bn1_b, bn1_m, bn1_v, hB);
    // ---- GAT layer 2 linear part: transpose W once, then WMMA GEMM ----
    transpose64<<<16, 256, 0, stream>>>(gat2_W, wt);
    launch_gemm(stream, hB, 64, wt, 64, nullptr, hA, 64, NN, 64, 64, 0);
    gat_scores<<<(NN + TPB - 1) / TPB, TPB, 0, stream>>>(hA, gat2_as, gat2_ad, sas, sad);
    gat_aggr<<<(NN * 64 + TPB - 1) / TPB, TPB, 0, stream>>>(hA, sas, sad, gat2_b,
                                                            bn2_g, bn2_b, bn2_m, bn2_v, hB);
    // ---- pack [h2|x] -> emb [5760,1122] ----
    pack_emb<<<(NN * 66 + TPB - 1) / TPB, TPB, 0, stream>>>(hB, x, emb);

    // ---- BiLSTM layer 1: hoisted input GEMM (bias fused), then 90 recurrent steps
    for (int d = 0; d < 2; ++d)
        launch_gemm(stream, emb, D_IN1, l1_Wih + (size_t)d * GATE * D_IN1, D_IN1,
                    l1_b + (size_t)d * GATE, Zx + (size_t)d * NGR * GATE, GATE,
                    NGR, GATE, D_IN1, 0);
    zero_f32<<<(2 * 2 * BATCH * HID + TPB - 1) / TPB, TPB, 0, stream>>>(hst, 2 * 2 * BATCH * HID);
    for (int t = 0; t < SEQ; ++t) {
        lstm_step_gemm<<<dim3(16, 2), 256, 0, stream>>>(Zx, l1_Whh, hst, Zf, t);
        lstm_cell<<<(2 * BATCH * HID + TPB - 1) / TPB, TPB, 0, stream>>>(Zf, hst, cst, ys1, t);
    }

    // ---- BiLSTM layer 2 ----
    for (int d = 0; d < 2; ++d)
        launch_gemm(stream, ys1, D_IN2, l2_Wih + (size_t)d * GATE * D_IN2, D_IN2,
                    l2_b + (size_t)d * GATE, Zx + (size_t)d * NGR * GATE, GATE,
                    NGR, GATE, D_IN2, 0);
    zero_f32<<<(2 * 2 * BATCH * HID + TPB - 1) / TPB, TPB, 0, stream>>>(hst, 2 * 2 * BATCH * HID);
    for (int t = 0; t < SEQ; ++t) {
        lstm_step_gemm<<<dim3(16, 2), 256, 0, stream>>>(Zx, l2_Whh, hst, Zf, t);
        lstm_cell<<<(2 * BATCH * HID + TPB - 1) / TPB, TPB, 0, stream>>>(Zf, hst, cst, ys2, t);
    }

    // ---- head ----
    launch_gemm(stream, ys2, 1024, hW1, 1024, hb1, hd1, 256, NGR, 256, 1024, 1);
    launch_gemm(stream, hd1, 256,  hW2, 256,  hb2, out, 57,  NGR, 57,  256,  0);
}